// WriteHead_59639915872656
// MI455X (gfx1250) — compile-verified
//
#include <hip/hip_runtime.h>
#include <hip/hip_bf16.h>
#include <math.h>

// ---------------- problem constants ----------------
#define BB   512      // batch
#define HH   512      // hidden
#define RR   2048     // rows
#define CC   64       // cols
#define MM   3        // shift half-width
#define OUTD 202      // 3*C + 2*M + 4
#define ZLD  208      // padded leading dim for z / params
#define EPS_COS   1e-8f
#define EPS_SHARP 1e-16f

typedef __attribute__((ext_vector_type(2))) float v2f;
typedef __attribute__((ext_vector_type(8))) float v8f;

// params layout per batch (stride ZLD floats):
//  [0..63]   k (tanh)
//  [64..127] e (sigmoid)
//  [128..191]a (tanh)
//  [192..198]s (softmax, 7)
//  [199] beta   [200] g   [201] gamma   [202] k_norm

// ---------------------------------------------------------------------------
// Kernel 1: z = h @ fc_w^T + fc_b via V_WMMA_F32_16X16X4_F32.
// One wave per 16x16 tile of z. grid = (M/16=32, ceil(202/16)=13), block = 32.
// A(16x4) layout: lane l holds row m=l&15; vgpr v = A[m][k + v + 2*(l>>4)]
// B(4x16) layout: vgpr v = B[k + v + 2*(l>>4)][l&15]   (B[k][n] = fc_w[n][k])
// C(16x16): vgpr v: rows v (lanes 0-15) / v+8 (lanes 16-31), col = l&15
// ---------------------------------------------------------------------------
__global__ __launch_bounds__(32) void gemm_z_wmma(const float* __restrict__ h,
                                                  const float* __restrict__ fc_w,
                                                  const float* __restrict__ fc_b,
                                                  float* __restrict__ z) {
  const int tm = blockIdx.x;          // M tile
  const int tn = blockIdx.y;          // N tile
  const int lane = threadIdx.x;       // 0..31
  const int lo = lane & 15;
  const int hi = lane >> 4;

  const int n  = tn * 16 + lo;
  const int ncl = (n < OUTD) ? n : (OUTD - 1);     // clamp reads, mask store

  const float* __restrict__ arow = h    + (size_t)(tm * 16 + lo) * HH;
  const float* __restrict__ brow = fc_w + (size_t)ncl * HH;

  v8f acc = {};
  for (int k = 0; k < HH; k += 4) {
    v2f a, b;
    a.x = arow[k + 2 * hi];
    a.y = arow[k + 2 * hi + 1];
    b.x = brow[k + 2 * hi];
    b.y = brow[k + 2 * hi + 1];
    acc = __builtin_amdgcn_wmma_f32_16x16x4_f32(
        /*neg_a=*/false, a, /*neg_b=*/false, b,
        /*c_mod=*/(short)0, acc, /*reuse_a=*/false, /*reuse_b=*/false);
  }

  if (n < OUTD) {
    const float bias = fc_b[n];
#pragma unroll
    for (int v = 0; v < 8; ++v) {
      const int row = tm * 16 + v + 8 * hi;
      z[(size_t)row * ZLD + n] = acc[v] + bias;
    }
  }
}

__device__ __forceinline__ float softplusf(float x) {
  return (x > 0.0f) ? (x + log1pf(expf(-x))) : log1pf(expf(x));
}
__device__ __forceinline__ float sigmoidf_(float x) {
  return 1.0f / (1.0f + expf(-x));
}

// ---------------------------------------------------------------------------
// Kernel 2: per-batch activations + ||k||.  grid = 512, block = 64.
// ---------------------------------------------------------------------------
__global__ __launch_bounds__(64) void activ_kernel(const float* __restrict__ z,
                                                   float* __restrict__ params) {
  const int b = blockIdx.x;
  const int t = threadIdx.x;  // 0..63
  const float* __restrict__ zb = z + (size_t)b * ZLD;
  float* __restrict__ pb = params + (size_t)b * ZLD;

  const float kv = tanhf(zb[t]);
  pb[t]        = kv;                       // k
  pb[64 + t]   = sigmoidf_(zb[74 + t]);    // e
  pb[128 + t]  = tanhf(zb[138 + t]);       // a

  __shared__ float red[64];
  red[t] = kv * kv;
  __syncthreads();
#pragma unroll
  for (int s = 32; s > 0; s >>= 1) {
    if (t < s) red[t] += red[t + s];
    __syncthreads();
  }
  if (t == 0) {
    float kn = sqrtf(red[0]);
    pb[202] = fmaxf(kn, EPS_COS);
    pb[199] = softplusf(zb[64]);                  // beta
    pb[200] = sigmoidf_(zb[65]);                  // g
    pb[201] = softplusf(zb[73]) + 1.0f;           // gamma
    float sv[7], mx = -3.4e38f;
#pragma unroll
    for (int j = 0; j < 7; ++j) { sv[j] = zb[66 + j]; mx = fmaxf(mx, sv[j]); }
    float sum = 0.0f;
#pragma unroll
    for (int j = 0; j < 7; ++j) { sv[j] = expf(sv[j] - mx); sum += sv[j]; }
#pragma unroll
    for (int j = 0; j < 7; ++j) pb[192 + j] = sv[j] / sum;
  }
}

// ---------------------------------------------------------------------------
// Kernel 3: cosine-similarity logits. One wave32 per memory row (64 floats:
// 2 per lane, coalesced), wave reduction via shfl_xor. 8 waves per block.
// ---------------------------------------------------------------------------
__global__ __launch_bounds__(256) void cosine_kernel(const float* __restrict__ mem,
                                                     const float* __restrict__ params,
                                                     float* __restrict__ logits) {
  const int wave = threadIdx.x >> 5;
  const int lane = threadIdx.x & 31;
  const long long row = (long long)blockIdx.x * 8 + wave;  // 0 .. B*R-1
  const int b = (int)(row >> 11);

  const float* __restrict__ m0 = mem + row * (long long)CC;
  const float* __restrict__ pb = params + (size_t)b * ZLD;

  const float k0 = pb[lane], k1 = pb[lane + 32];
  const float x0 = m0[lane], x1 = m0[lane + 32];
  float num = k0 * x0 + k1 * x1;
  float nrm = x0 * x0 + x1 * x1;
#pragma unroll
  for (int off = 16; off > 0; off >>= 1) {
    num += __shfl_xor(num, off, 32);
    nrm += __shfl_xor(nrm, off, 32);
  }
  if (lane == 0) {
    const float mn = fmaxf(sqrtf(nrm), EPS_COS);
    logits[row] = pb[199] * (num / (pb[202] * mn));
  }
}

// ---------------------------------------------------------------------------
// Kernel 4: per-batch softmax over R, gate, 7-tap circular conv, sharpen,
// normalize -> w. One 512-thread block per batch; whole row in LDS.
// ---------------------------------------------------------------------------
__global__ __launch_bounds__(512) void focus_kernel(const float* __restrict__ logits,
                                                    const float* __restrict__ prev_w,
                                                    const float* __restrict__ params,
                                                    float* __restrict__ w_out) {
  const int b = blockIdx.x;
  const int t = threadIdx.x;  // 0..511, each owns 4 strided elements
  __shared__ float wg[RR];
  __shared__ float red[512];

  const float* __restrict__ lg = logits + (long long)b * RR;
  const float* __restrict__ pw = prev_w + (long long)b * RR;
  const float* __restrict__ pb = params + (size_t)b * ZLD;

  float v[4];
  float mx = -3.4e38f;
#pragma unroll
  for (int i = 0; i < 4; ++i) { v[i] = lg[t + 512 * i]; mx = fmaxf(mx, v[i]); }
  red[t] = mx; __syncthreads();
  for (int s = 256; s > 0; s >>= 1) { if (t < s) red[t] = fmaxf(red[t], red[t + s]); __syncthreads(); }
  mx = red[0]; __syncthreads();

  float psum = 0.0f;
#pragma unroll
  for (int i = 0; i < 4; ++i) { v[i] = expf(v[i] - mx); psum += v[i]; }
  red[t] = psum; __syncthreads();
  for (int s = 256; s > 0; s >>= 1) { if (t < s) red[t] += red[t + s]; __syncthreads(); }
  const float tot = red[0]; __syncthreads();

  const float g = pb[200];
#pragma unroll
  for (int i = 0; i < 4; ++i) {
    const int r = t + 512 * i;
    wg[r] = g * (v[i] / tot) + (1.0f - g) * pw[r];
  }
  __syncthreads();

  float sh[7];
#pragma unroll
  for (int j = 0; j < 7; ++j) sh[j] = pb[192 + j];
  const float gamma = pb[201];

  float p[4];
  float ps = 0.0f;
#pragma unroll
  for (int i = 0; i < 4; ++i) {
    const int r = t + 512 * i;
    float acc = 0.0f;
#pragma unroll
    for (int j = 0; j < 7; ++j) acc += sh[j] * wg[(r + j - MM) & (RR - 1)];
    const float pwr = powf(acc, gamma);
    p[i] = pwr; ps += pwr;
  }
  red[t] = ps; __syncthreads();
  for (int s = 256; s > 0; s >>= 1) { if (t < s) red[t] += red[t + s]; __syncthreads(); }
  const float inv = 1.0f / (red[0] + EPS_SHARP);
#pragma unroll
  for (int i = 0; i < 4; ++i) w_out[(long long)b * RR + t + 512 * i] = p[i] * inv;
}

// ---------------------------------------------------------------------------
// Kernel 5: new_mem = mem*(1 - w*e) + w*a.  float4 per thread, coalesced.
// total float4 elements = B*R*C/4 = 16,777,216.
// ---------------------------------------------------------------------------
__global__ __launch_bounds__(256) void writemem_kernel(const float* __restrict__ mem,
                                                       const float* __restrict__ params,
                                                       const float* __restrict__ w,
                                                       float* __restrict__ out) {
  const long long t = (long long)blockIdx.x * 256 + threadIdx.x;  // float4 idx
  const long long row = t >> 4;          // b*R + r
  const int c4 = (int)(t & 15);
  const int b = (int)(row >> 11);

  const float4 mv = ((const float4*)mem)[t];
  const float* __restrict__ pb = params + (size_t)b * ZLD;
  const float wv = w[row];

  const float4 ev = *(const float4*)(pb + 64 + c4 * 4);
  const float4 av = *(const float4*)(pb + 128 + c4 * 4);

  float4 o;
  o.x = mv.x * (1.0f - wv * ev.x) + wv * av.x;
  o.y = mv.y * (1.0f - wv * ev.y) + wv * av.y;
  o.z = mv.z * (1.0f - wv * ev.z) + wv * av.z;
  o.w = mv.w * (1.0f - wv * ev.w) + wv * av.w;
  ((float4*)out)[t] = o;
}

// ---------------------------------------------------------------------------
extern "C" void kernel_launch(void* const* d_in, const int* in_sizes, int n_in,
                              void* d_out, int out_size, void* d_ws, size_t ws_size,
                              hipStream_t stream) {
  (void)in_sizes; (void)n_in; (void)out_size; (void)ws_size;
  const float* h      = (const float*)d_in[0];   // 512*512
  const float* prev_w = (const float*)d_in[1];   // 512*2048
  const float* memory = (const float*)d_in[2];   // 512*2048*64
  const float* fc_w   = (const float*)d_in[3];   // 202*512
  const float* fc_b   = (const float*)d_in[4];   // 202

  float* out_w   = (float*)d_out;                          // [512*2048]
  float* out_mem = out_w + (size_t)BB * RR;                // [512*2048*64]

  // Scratch: logits + z live inside the (not-yet-written) new_mem region of
  // d_out; both are fully consumed before writemem_kernel overwrites it.
  float* logits = out_mem;                                  // B*R floats
  float* zbuf   = out_mem + (size_t)2 * 1024 * 1024;        // B*ZLD floats
  // params must survive until writemem_kernel -> keep in ws (426 KB).
  float* params = (float*)d_ws;                             // B*ZLD floats

  dim3 gGemm(BB / 16, (OUTD + 15) / 16);
  gemm_z_wmma<<<gGemm, 32, 0, stream>>>(h, fc_w, fc_b, zbuf);

  activ_kernel<<<BB, 64, 0, stream>>>(zbuf, params);

  cosine_kernel<<<(BB * RR) / 8, 256, 0, stream>>>(memory, params, logits);

  focus_kernel<<<BB, 512, 0, stream>>>(logits, prev_w, params, out_w);

  writemem_kernel<<<(BB * (long long)RR * CC / 4) / 256, 256, 0, stream>>>(
      memory, params, out_w, out_mem);
}